// LookUpGCN_19215683682347
// MI455X (gfx1250) — compile-verified
//
#include <hip/hip_runtime.h>
#include <math.h>

// LookUpGCN for MI455X (gfx1250, wave32).
// Dense x@W uses V_WMMA_F32_16X16X4_F32 (fp32 exact). Edge aggregation is
// L2-resident float atomics. LayerNorm via wave32 shuffle reduction.

#define NN      50000
#define NE      600000
#define DIMV    128
#define LN_EPS  1e-5f

typedef float v2f __attribute__((ext_vector_type(2)));
typedef float v8f __attribute__((ext_vector_type(8)));

// ---------------- degree / normalization ----------------

__global__ void k_deg_init(float* __restrict__ deg) {
  int i = blockIdx.x * blockDim.x + threadIdx.x;
  if (i < NN) deg[i] = 1.0f;               // self-loop weight
}

__global__ void k_deg_edges(const int* __restrict__ col,
                            const float* __restrict__ ew,
                            float* __restrict__ deg) {
  int e = blockIdx.x * blockDim.x + threadIdx.x;
  if (e < NE) atomicAdd(&deg[col[e]], ew[e]);
}

__global__ void k_dis(const float* __restrict__ deg, float* __restrict__ dis) {
  int i = blockIdx.x * blockDim.x + threadIdx.x;
  if (i < NN) {
    float d = deg[i];
    dis[i] = (d > 0.f) ? rsqrtf(fmaxf(d, 1e-12f)) : 0.f;
  }
}

// ---------------- embedding gather ----------------

__global__ void k_gather(const int* __restrict__ ids,
                         const float* __restrict__ emb,
                         float* __restrict__ x) {
  int idx = blockIdx.x * blockDim.x + threadIdx.x;   // over NN*DIMV
  if (idx < NN * DIMV) {
    int i = idx >> 7;
    int d = idx & (DIMV - 1);
    x[idx] = emb[(size_t)ids[i] * DIMV + d];
  }
}

// ---------------- dense GEMM: H = X @ W  via v_wmma_f32_16x16x4_f32 ----------
// Block = 256 threads = 8 waves. Block b owns rows [16b,16b+16); wave w owns
// output columns [16w,16w+16). K=128 consumed in 32 steps of K=4.
// A frag (16x4 f32): lane<16 -> row=lane, K=k0..k0+1 ; lane>=16 -> K=k0+2..k0+3
// B frag (4x16 f32): mirrored; lane holds W[k][n0+lane%16] pairs.
// D (16x16 f32): VGPR r -> row r (lanes 0-15) / row 8+r (lanes 16-31).

__global__ void k_gemm_xW(const float* __restrict__ X,
                          const float* __restrict__ W,
                          float* __restrict__ H) {
  const int lane  = threadIdx.x & 31;
  const int wave  = threadIdx.x >> 5;      // 0..7 -> N tile
  const int row0  = blockIdx.x * 16;       // 3125 blocks, exact fit
  const int m     = lane & 15;
  const int khalf = lane >> 4;             // 0 or 1
  const int n0    = wave * 16;

  const float* __restrict__ xrow = X + (size_t)(row0 + m) * DIMV;
  v8f acc = {};

#pragma unroll
  for (int k0 = 0; k0 < DIMV; k0 += 4) {
    const int ka = k0 + khalf * 2;
    v2f a;
    a.x = xrow[ka];
    a.y = xrow[ka + 1];
    v2f b;
    b.x = W[(size_t)ka * DIMV + n0 + m];
    b.y = W[(size_t)(ka + 1) * DIMV + n0 + m];
    acc = __builtin_amdgcn_wmma_f32_16x16x4_f32(
        /*neg_a=*/false, a, /*neg_b=*/false, b,
        /*c_mod=*/(short)0, acc, /*reuse_a=*/false, /*reuse_b=*/false);
  }

  const int outm = row0 + (khalf ? 8 : 0);
  float* hp = H + (size_t)outm * DIMV + n0 + m;
#pragma unroll
  for (int r = 0; r < 8; ++r) hp[(size_t)r * DIMV] = acc[r];
}

// ---------------- self-loop init: agg[i] = dis[i]^2 * h[i] ------------------

__global__ void k_init_agg(const float* __restrict__ H,
                           const float* __restrict__ dis,
                           float* __restrict__ AGG) {
  int idx = blockIdx.x * blockDim.x + threadIdx.x;   // over NN*DIMV
  if (idx < NN * DIMV) {
    int i = idx >> 7;
    float di = dis[i];
    AGG[idx] = di * di * H[idx];
  }
}

// ---------------- edge messages: one wave per edge --------------------------

__global__ void k_edge_msgs(const int* __restrict__ row,
                            const int* __restrict__ col,
                            const float* __restrict__ ew,
                            const float* __restrict__ dis,
                            const float* __restrict__ H,
                            float* __restrict__ AGG) {
  const int lane = threadIdx.x & 31;
  const int e = blockIdx.x * 8 + (threadIdx.x >> 5);
  if (e >= NE) return;
  const int s = row[e];
  const int t = col[e];
  const float coef = dis[s] * ew[e] * dis[t];

  const float4 hv = ((const float4*)(H + (size_t)s * DIMV))[lane];
  float* ap = AGG + (size_t)t * DIMV + lane * 4;
  atomicAdd(ap + 0, coef * hv.x);
  atomicAdd(ap + 1, coef * hv.y);
  atomicAdd(ap + 2, coef * hv.z);
  atomicAdd(ap + 3, coef * hv.w);
}

// ---------------- residual + bias + LayerNorm (one wave per row) ------------

__global__ void k_post_ln(const float* __restrict__ X,
                          const float* __restrict__ AGG,
                          const float* __restrict__ bias,
                          const float* __restrict__ gamma,
                          const float* __restrict__ beta,
                          float* __restrict__ OUT) {
  const int lane = threadIdx.x & 31;
  const int i = blockIdx.x * 8 + (threadIdx.x >> 5);
  if (i >= NN) return;

  const float4 xv = ((const float4*)(X + (size_t)i * DIMV))[lane];
  const float4 av = ((const float4*)(AGG + (size_t)i * DIMV))[lane];
  const float4 bv = ((const float4*)bias)[lane];

  float v0 = xv.x + av.x + bv.x;
  float v1 = xv.y + av.y + bv.y;
  float v2 = xv.z + av.z + bv.z;
  float v3 = xv.w + av.w + bv.w;

  float s = v0 + v1 + v2 + v3;
#pragma unroll
  for (int off = 16; off > 0; off >>= 1) s += __shfl_xor(s, off, 32);
  const float mu = s * (1.0f / DIMV);

  const float d0 = v0 - mu, d1 = v1 - mu, d2 = v2 - mu, d3 = v3 - mu;
  float q = d0 * d0 + d1 * d1 + d2 * d2 + d3 * d3;
#pragma unroll
  for (int off = 16; off > 0; off >>= 1) q += __shfl_xor(q, off, 32);
  const float inv = rsqrtf(q * (1.0f / DIMV) + LN_EPS);

  const float4 gv = ((const float4*)gamma)[lane];
  const float4 ev = ((const float4*)beta)[lane];
  float4 o;
  o.x = d0 * inv * gv.x + ev.x;
  o.y = d1 * inv * gv.y + ev.y;
  o.z = d2 * inv * gv.z + ev.z;
  o.w = d3 * inv * gv.w + ev.w;
  ((float4*)(OUT + (size_t)i * DIMV))[lane] = o;
}

// ---------------- launch ----------------------------------------------------

extern "C" void kernel_launch(void* const* d_in, const int* in_sizes, int n_in,
                              void* d_out, int out_size, void* d_ws, size_t ws_size,
                              hipStream_t stream) {
  const int*   node_ids = (const int*)d_in[0];
  const int*   erow     = (const int*)d_in[1];          // edge_index[0]
  const int*   ecol     = ((const int*)d_in[1]) + NE;   // edge_index[1]
  const float* ew       = (const float*)d_in[2];
  const float* emb      = (const float*)d_in[3];
  const float* W1       = (const float*)d_in[4];
  const float* b1       = (const float*)d_in[5];
  const float* W2       = (const float*)d_in[6];
  const float* b2       = (const float*)d_in[7];
  const float* g1       = (const float*)d_in[8];
  const float* be1      = (const float*)d_in[9];
  const float* g2       = (const float*)d_in[10];
  const float* be2      = (const float*)d_in[11];
  float* out = (float*)d_out;

  float* ws  = (float*)d_ws;
  float* x   = ws;                              // 6.4M floats
  float* h   = ws + (size_t)NN * DIMV;          // 6.4M floats
  float* agg = ws + (size_t)2 * NN * DIMV;      // 6.4M floats
  float* deg = ws + (size_t)3 * NN * DIMV;      // 50k floats
  float* dis = deg + NN;                        // 50k floats

  const int B = 256;
  const int gNode  = (NN + B - 1) / B;          // 196
  const int gEdge  = (NE + B - 1) / B;          // 2344
  const int gFeat  = (NN * DIMV + B - 1) / B;   // 25000
  const int gGemm  = NN / 16;                   // 3125 (exact)
  const int gEMsg  = (NE + 7) / 8;              // 75000 (8 edges/block)
  const int gRow   = (NN + 7) / 8;              // 6250  (8 rows/block)

  // normalization coefficients (recomputed every call: determinism)
  k_deg_init <<<gNode, B, 0, stream>>>(deg);
  k_deg_edges<<<gEdge, B, 0, stream>>>(ecol, ew, deg);
  k_dis      <<<gNode, B, 0, stream>>>(deg, dis);

  // x = emb[node_ids]
  k_gather   <<<gFeat, B, 0, stream>>>(node_ids, emb, x);

  // ---- layer 1 ----
  k_gemm_xW  <<<gGemm, B, 0, stream>>>(x, W1, h);
  k_init_agg <<<gFeat, B, 0, stream>>>(h, dis, agg);
  k_edge_msgs<<<gEMsg, B, 0, stream>>>(erow, ecol, ew, dis, h, agg);
  k_post_ln  <<<gRow,  B, 0, stream>>>(x, agg, b1, g1, be1, x);   // in-place update of x

  // ---- layer 2 ----
  k_gemm_xW  <<<gGemm, B, 0, stream>>>(x, W2, h);
  k_init_agg <<<gFeat, B, 0, stream>>>(h, dis, agg);
  k_edge_msgs<<<gEMsg, B, 0, stream>>>(erow, ecol, ew, dis, h, agg);
  k_post_ln  <<<gRow,  B, 0, stream>>>(x, agg, b2, g2, be2, out);
}